// GMMD_75256416960849
// MI455X (gfx1250) — compile-verified
//
#include <hip/hip_runtime.h>
#include <math.h>

typedef __attribute__((ext_vector_type(2))) float v2f;
typedef __attribute__((ext_vector_type(8))) float v8f;

constexpr int N     = 8192;
constexpr int D     = 256;
constexpr int TWO_N = 16384;
constexpr int BLK   = 128;          // output block edge (8x8 wmma tiles)
constexpr int KC    = 32;           // K chunk staged through LDS
constexpr int LSTR  = 36;           // LDS row stride in floats (bank-conflict-free, 144B = 16B aligned)
constexpr int NB    = TWO_N / BLK;  // 128 block rows/cols

// ---------------------------------------------------------------------------
// Phase 0: G[r] = gated feature row, norms[r] = ||G[r]||^2
// rows 0..N-1 -> feat_high * gate ; rows N..2N-1 -> feat_low * gate
// ---------------------------------------------------------------------------
__global__ __launch_bounds__(256)
void gmmd_gate_norm(const float* __restrict__ fh,
                    const float* __restrict__ fl,
                    const float* __restrict__ gw,
                    float* __restrict__ G,
                    float* __restrict__ norms) {
    const int row = blockIdx.x;     // 0..16383
    const int t   = threadIdx.x;    // 0..255 == D
    const int src_row = (row < N) ? row : (row - N);
    const float* src = (row < N) ? (fh + (size_t)src_row * D)
                                 : (fl + (size_t)src_row * D);
    const float g = src[t] * gw[(size_t)src_row * D + t];
    G[(size_t)row * D + t] = g;

    __shared__ float red[256];
    red[t] = g * g;
    __syncthreads();
    for (int s = 128; s > 0; s >>= 1) {
        if (t < s) red[t] += red[t + s];
        __syncthreads();
    }
    if (t == 0) norms[row] = red[0];
}

// ---------------------------------------------------------------------------
// Phase 1: signed exp-sum of Gram blocks via V_WMMA_F32_16X16X4_F32.
// Grid: (NB, NB); only bi <= bj blocks compute (symmetry, off-diag weight 2).
// Workgroup = 8 waves; wave w computes the 16x128 strip (tile row w).
// Register double-buffered global->LDS staging + batched fragment loads.
// ---------------------------------------------------------------------------
__global__ __launch_bounds__(256)
void gmmd_gram(const float* __restrict__ G,
               const float* __restrict__ norms,
               float* __restrict__ partials) {
    const int bj  = blockIdx.x;
    const int bi  = blockIdx.y;
    const int bid = bi * NB + bj;

    if (bi > bj) {                       // lower triangle: covered by symmetry
        if (threadIdx.x == 0) partials[bid] = 0.0f;
        return;
    }

    __shared__ __align__(16) float As[BLK * LSTR];
    __shared__ __align__(16) float Bs[BLK * LSTR];
    __shared__ float wred[8];

    const int t    = threadIdx.x;
    const int wave = t >> 5;             // 0..7
    const int lane = t & 31;

    const int rowBase = bi * BLK;
    const int colBase = bj * BLK;

    // this thread's 4 staging slots: 1024 float4 slots per matrix per chunk
    // slot idx = t + i*256 ; row = idx>>3 ; float4-within-row = idx&7
    const int r0  = t >> 3;              // rows for i-th slot: r0 + i*32
    const int c4f = (t & 7) * 4;         // float offset within the 32-float row

    v8f acc[8];
#pragma unroll
    for (int i = 0; i < 8; ++i) { v8f z = {}; acc[i] = z; }

    // fragment addressing (f32 16x16x4):
    //  A: lane L holds row m = L&15, K = {kh, kh+1} with kh = (L>>4)*2 (VGPR0/1)
    //  B: same pattern over columns (B rows are rows of G, Gram is G*G^T)
    const int m  = lane & 15;
    const int kh = (lane >> 4) * 2;

    // ---- prologue: prefetch k-chunk 0 into registers --------------------
    float4 ra[4], rb[4];
#pragma unroll
    for (int i = 0; i < 4; ++i) {
        const int r = r0 + i * 32;
        ra[i] = *(const float4*)(G + (size_t)(rowBase + r) * D + c4f);
        rb[i] = *(const float4*)(G + (size_t)(colBase + r) * D + c4f);
    }

    for (int k0 = 0; k0 < D; k0 += KC) {
        // ---- commit staged registers to LDS -----------------------------
#pragma unroll
        for (int i = 0; i < 4; ++i) {
            const int r = r0 + i * 32;
            *(float4*)(&As[r * LSTR + c4f]) = ra[i];
            *(float4*)(&Bs[r * LSTR + c4f]) = rb[i];
        }
        __syncthreads();

        // ---- prefetch next chunk (overlaps with WMMA compute below) -----
        if (k0 + KC < D) {
#pragma unroll
            for (int i = 0; i < 4; ++i) {
                const int r = r0 + i * 32;
                ra[i] = *(const float4*)(G + (size_t)(rowBase + r) * D + (k0 + KC) + c4f);
                rb[i] = *(const float4*)(G + (size_t)(colBase + r) * D + (k0 + KC) + c4f);
            }
        }

        // ---- compute: batch all fragment loads, then WMMA burst ---------
#pragma unroll
        for (int kk = 0; kk < KC; kk += 4) {
            v2f a;
            {
                const float* ap = &As[(wave * 16 + m) * LSTR + kk + kh];
                a.x = ap[0]; a.y = ap[1];
            }
            v2f b[8];
#pragma unroll
            for (int tc = 0; tc < 8; ++tc) {
                const float* bp = &Bs[(tc * 16 + m) * LSTR + kk + kh];
                b[tc].x = bp[0]; b[tc].y = bp[1];
            }
#pragma unroll
            for (int tc = 0; tc < 8; ++tc) {
                acc[tc] = __builtin_amdgcn_wmma_f32_16x16x4_f32(
                    false, a, false, b[tc], (short)0, acc[tc], false, false);
            }
        }
        __syncthreads();
    }

    // Epilogue. C/D layout: VGPR v -> M = v + 8*(lane>>4), N = lane&15.
    const int nn   = lane & 15;
    const int mofs = (lane >> 4) * 8;

    float nrow[8];
#pragma unroll
    for (int v = 0; v < 8; ++v)
        nrow[v] = norms[rowBase + wave * 16 + mofs + v];

    float wsum = 0.0f;
#pragma unroll
    for (int tc = 0; tc < 8; ++tc) {
        const float ncol = norms[colBase + tc * 16 + nn];
#pragma unroll
        for (int v = 0; v < 8; ++v) {
            const float dot = acc[tc][v];
            wsum += __expf(dot - 0.5f * (nrow[v] + ncol));
        }
    }

    // block-uniform sign and symmetry weight
    const float srow = (rowBase < N) ? 1.0f : -1.0f;
    const float scol = (colBase < N) ? 1.0f : -1.0f;
    wsum *= srow * scol * ((bi == bj) ? 1.0f : 2.0f);

    // wave32 reduce, then cross-wave via LDS
    for (int off = 16; off > 0; off >>= 1)
        wsum += __shfl_down(wsum, off, 32);
    if (lane == 0) wred[wave] = wsum;
    __syncthreads();
    if (t == 0) {
        float s = 0.0f;
#pragma unroll
        for (int i = 0; i < 8; ++i) s += wred[i];
        partials[bid] = s;
    }
}

// ---------------------------------------------------------------------------
// Phase 2: deterministic single-workgroup reduce of NB*NB partials.
// ---------------------------------------------------------------------------
__global__ __launch_bounds__(256)
void gmmd_reduce(const float* __restrict__ partials, int n,
                 float* __restrict__ out) {
    __shared__ float red[256];
    float s = 0.0f;
    for (int i = threadIdx.x; i < n; i += 256) s += partials[i];
    red[threadIdx.x] = s;
    __syncthreads();
    for (int k = 128; k > 0; k >>= 1) {
        if (threadIdx.x < k) red[threadIdx.x] += red[threadIdx.x + k];
        __syncthreads();
    }
    if (threadIdx.x == 0)
        out[0] = red[0] * (1.0f / ((float)N * (float)N)); // exact pow2 scale
}

extern "C" void kernel_launch(void* const* d_in, const int* in_sizes, int n_in,
                              void* d_out, int out_size, void* d_ws, size_t ws_size,
                              hipStream_t stream) {
    const float* fh = (const float*)d_in[0];   // feat_high  [N, D]
    const float* fl = (const float*)d_in[1];   // feat_low   [N, D]
    const float* gw = (const float*)d_in[2];   // gate_weight[N, D]

    float* G        = (float*)d_ws;                 // 2N*D floats (16 MB)
    float* norms    = G + (size_t)TWO_N * D;        // 2N floats
    float* partials = norms + TWO_N;                // NB*NB floats

    gmmd_gate_norm<<<TWO_N, 256, 0, stream>>>(fh, fl, gw, G, norms);

    dim3 grid(NB, NB);
    gmmd_gram<<<grid, 256, 0, stream>>>(G, norms, partials);

    gmmd_reduce<<<1, 256, 0, stream>>>(partials, NB * NB, (float*)d_out);
}